// CFA_44083544326661
// MI455X (gfx1250) — compile-verified
//
#include <hip/hip_runtime.h>
#include <hip/hip_bf16.h>

// Problem constants: tensor [N, D] f32, N=8192, D=256.
#define N_ROWS 8192
#define D_DIM  256
#define EPSV   1e-8f

typedef __attribute__((ext_vector_type(2))) float v2f;
typedef __attribute__((ext_vector_type(8))) float v8f;

__device__ __forceinline__ float wave_sum32(float v) {
    #pragma unroll
    for (int off = 16; off > 0; off >>= 1)
        v += __shfl_xor(v, off, 32);
    return v;
}

// Zero G (65536 floats) and s (256 floats), laid out contiguously in ws.
__global__ __launch_bounds__(256) void cfa_zero(float* __restrict__ p, int n) {
    int i = blockIdx.x * 256 + threadIdx.x;
    if (i < n) p[i] = 0.0f;
}

// One wave (32 lanes) per row: x_hat = row / max(||row||, eps); accumulate s = sum of x_hat rows.
__global__ __launch_bounds__(256) void cfa_normalize(const float* __restrict__ X,
                                                     float* __restrict__ xhat,
                                                     float* __restrict__ s) {
    __shared__ float sblk[D_DIM];
    int tid  = threadIdx.x;
    sblk[tid] = 0.0f;
    __syncthreads();

    int lane = tid & 31;
    int row  = blockIdx.x * 8 + (tid >> 5);
    const float* xr = X + (size_t)row * D_DIM + lane * 8;
    float4 v0 = *(const float4*)(xr);
    float4 v1 = *(const float4*)(xr + 4);

    float ss = v0.x*v0.x + v0.y*v0.y + v0.z*v0.z + v0.w*v0.w
             + v1.x*v1.x + v1.y*v1.y + v1.z*v1.z + v1.w*v1.w;
    ss = wave_sum32(ss);
    float inv = 1.0f / fmaxf(sqrtf(ss), EPSV);

    v0.x *= inv; v0.y *= inv; v0.z *= inv; v0.w *= inv;
    v1.x *= inv; v1.y *= inv; v1.z *= inv; v1.w *= inv;

    float* xo = xhat + (size_t)row * D_DIM + lane * 8;
    *(float4*)(xo)     = v0;
    *(float4*)(xo + 4) = v1;

    int c = lane * 8;
    atomicAdd(&sblk[c + 0], v0.x); atomicAdd(&sblk[c + 1], v0.y);
    atomicAdd(&sblk[c + 2], v0.z); atomicAdd(&sblk[c + 3], v0.w);
    atomicAdd(&sblk[c + 4], v1.x); atomicAdd(&sblk[c + 5], v1.y);
    atomicAdd(&sblk[c + 6], v1.z); atomicAdd(&sblk[c + 7], v1.w);
    __syncthreads();
    atomicAdd(&s[tid], sblk[tid]);
}

// rs[i] = dot(x_hat[i], s)  (== row sum of the cosine-similarity matrix)
__global__ __launch_bounds__(256) void cfa_rowsum(const float* __restrict__ xhat,
                                                  const float* __restrict__ s,
                                                  float* __restrict__ rs) {
    int tid  = threadIdx.x;
    int lane = tid & 31;
    int row  = blockIdx.x * 8 + (tid >> 5);
    const float* xr = xhat + (size_t)row * D_DIM + lane * 8;
    const float* sr = s + lane * 8;
    float acc = 0.0f;
    #pragma unroll
    for (int j = 0; j < 8; ++j) acc += xr[j] * sr[j];
    acc = wave_sum32(acc);
    if (lane == 0) rs[row] = acc;
}

// GEMM1: G[256x256] += x_hat^T (256xN) @ X (NxD). Split-K: 16x16 output tiles x 8 K-chunks.
// One wave per (tile, chunk); accumulates with V_WMMA_F32_16X16X4_F32, commits via f32 atomics.
__global__ __launch_bounds__(256) void cfa_gemm1(const float* __restrict__ xhat,
                                                 const float* __restrict__ X,
                                                 float* __restrict__ G) {
    int wid   = blockIdx.x * 8 + (threadIdx.x >> 5);   // 0..2047
    int lane  = threadIdx.x & 31;
    int chunk = wid & 7;                                // K split
    int tile  = wid >> 3;                               // 0..255
    int tm    = (tile >> 4) << 4;                       // d1 base (rows of G)
    int tn    = (tile & 15) << 4;                       // d2 base (cols of G)

    int m  = lane & 15;                                 // A: M index / B: N index
    int kb = (lane >> 4) << 1;                          // upper lanes hold K+2

    v8f c = {};
    int kstart = chunk * (N_ROWS / 8);
    #pragma unroll 2
    for (int it = 0; it < (N_ROWS / 8) / 4; ++it) {
        int i = kstart + it * 4 + kb;
        v2f a, b;
        a.x = xhat[(size_t)i * D_DIM + tm + m];         // A[m][k]   = x_hat[i][tm+m]
        a.y = xhat[(size_t)(i + 1) * D_DIM + tm + m];   // A[m][k+1]
        b.x = X[(size_t)i * D_DIM + tn + m];            // B[k][n]   = X[i][tn+n]
        b.y = X[(size_t)(i + 1) * D_DIM + tn + m];      // B[k+1][n]
        c = __builtin_amdgcn_wmma_f32_16x16x4_f32(false, a, false, b,
                                                  (short)0, c, false, false);
    }

    int half = (lane >> 4) << 3;                        // C: upper lanes hold M+8
    int col  = tn + (lane & 15);
    #pragma unroll
    for (int v = 0; v < 8; ++v)
        atomicAdd(&G[(size_t)(tm + half + v) * D_DIM + col], c[v]);
}

// GEMM2: out = (x_hat @ G) / rs (row-wise). One wave per 16x16 output tile, K=256.
__global__ __launch_bounds__(256) void cfa_gemm2(const float* __restrict__ xhat,
                                                 const float* __restrict__ G,
                                                 const float* __restrict__ rs,
                                                 float* __restrict__ out) {
    int wid  = blockIdx.x * 8 + (threadIdx.x >> 5);     // 0..8191
    int lane = threadIdx.x & 31;
    int i0   = (wid >> 4) << 4;                         // row-tile base (512 tiles)
    int n0   = (wid & 15) << 4;                         // col-tile base (16 tiles)

    int m  = lane & 15;
    int kb = (lane >> 4) << 1;

    v8f c = {};
    const float* arow = xhat + (size_t)(i0 + m) * D_DIM + kb;  // contiguous K pair -> v2f load
    #pragma unroll 4
    for (int k0 = 0; k0 < D_DIM; k0 += 4) {
        v2f a = *(const v2f*)(arow + k0);               // A[m][k], A[m][k+1]
        v2f b;
        b.x = G[(size_t)(k0 + kb) * D_DIM + n0 + m];    // B[k][n]
        b.y = G[(size_t)(k0 + kb + 1) * D_DIM + n0 + m];
        c = __builtin_amdgcn_wmma_f32_16x16x4_f32(false, a, false, b,
                                                  (short)0, c, false, false);
    }

    int half = (lane >> 4) << 3;
    int col  = n0 + (lane & 15);
    #pragma unroll
    for (int v = 0; v < 8; ++v) {
        int row = i0 + half + v;
        out[(size_t)row * D_DIM + col] = c[v] / rs[row];   // coalesced 64B per half-wave
    }
}

extern "C" void kernel_launch(void* const* d_in, const int* in_sizes, int n_in,
                              void* d_out, int out_size, void* d_ws, size_t ws_size,
                              hipStream_t stream) {
    (void)in_sizes; (void)n_in; (void)out_size; (void)ws_size;
    const float* X  = (const float*)d_in[0];
    float* out      = (float*)d_out;

    // Workspace layout (floats): x_hat[N*D] | G[D*D] | s[D] | rs[N]  (~8.3 MB total)
    float* xhat = (float*)d_ws;
    float* G    = xhat + (size_t)N_ROWS * D_DIM;
    float* s    = G + D_DIM * D_DIM;
    float* rs   = s + D_DIM;

    // G and s are contiguous -> zero both with one launch.
    cfa_zero<<<(D_DIM * D_DIM + D_DIM + 255) / 256, 256, 0, stream>>>(G, D_DIM * D_DIM + D_DIM);
    cfa_normalize<<<N_ROWS / 8, 256, 0, stream>>>(X, xhat, s);
    cfa_rowsum   <<<N_ROWS / 8, 256, 0, stream>>>(xhat, s, rs);
    cfa_gemm1    <<<(16 * 16 * 8) / 8, 256, 0, stream>>>(xhat, X, G);     // 2048 waves
    cfa_gemm2    <<<(512 * 16) / 8, 256, 0, stream>>>(xhat, G, rs, out);  // 8192 waves
}